// GCNLayer_23716809408972
// MI455X (gfx1250) — compile-verified
//
#include <hip/hip_runtime.h>

#define GCN_D 128
#define CHUNK 8           // gathered rows per batch (8 x 512B = 4KB per buffer)
#define AGG_BLOCK 256     // 8 waves
#define AGG_WAVES (AGG_BLOCK / 32)

// ---------------------------------------------------------------------------
// gfx1250 async-copy helpers (assembled OK in round 2).
// global_load_async_to_lds_b128: VDST = per-lane LDS byte address (wave-
// relative; low 32 bits of a generic pointer to __shared__ per ISA 10.2),
// VADDR = per-lane 64-bit global address, SADDR = off. Tracked by ASYNCcnt;
// async loads complete IN ORDER, which enables threshold-wait pipelining.
// ---------------------------------------------------------------------------
__device__ __forceinline__ void async_gather_b128(unsigned lds_byte, const float* g) {
  unsigned long long ga = (unsigned long long)(size_t)g;
  asm volatile("global_load_async_to_lds_b128 %0, %1, off"
               :: "v"(lds_byte), "v"(ga)
               : "memory");
}

template <int N>
__device__ __forceinline__ void wait_async_le() {
#if __has_builtin(__builtin_amdgcn_s_wait_asynccnt)
  __builtin_amdgcn_s_wait_asynccnt(N);
#else
  asm volatile("s_wait_asynccnt %0" :: "n"(N) : "memory");
#endif
}

// ---------------------------------------------------------------------------
// issue m async row gathers for edges [j, j+m) into one LDS buffer
// ---------------------------------------------------------------------------
__device__ __forceinline__ void issue_chunk(const int* __restrict__ edge_src, int j, int m,
                                            const float* __restrict__ src_emb,
                                            const float* __restrict__ dst_emb, int n_src,
                                            unsigned lds_base, int lane) {
  __builtin_prefetch(edge_src + j + CHUNK, 0, 0);  // next batch of src ids
  for (int c = 0; c < m; ++c) {
    const int s = edge_src[j + c];  // uniform -> s_load
    const float* row = (s < n_src) ? (src_emb + (size_t)s * GCN_D)
                                   : (dst_emb + (size_t)(s - n_src) * GCN_D);
    async_gather_b128(lds_base + (unsigned)(c * (GCN_D * 4)), row + lane * 4);
  }
}

// ---------------------------------------------------------------------------
// accumulate m staged rows from one LDS buffer with per-source normalization
// ---------------------------------------------------------------------------
__device__ __forceinline__ void consume_chunk(const int* __restrict__ edge_src, int j, int m,
                                              const float* __restrict__ scale_out,
                                              const float* stage_buf, int lane, float4& acc) {
  for (int c = 0; c < m; ++c) {
    const int s = edge_src[j + c];
    const float sc = scale_out[s];  // uniform -> s_load
    float4 v = *(const float4*)(stage_buf + c * GCN_D + lane * 4);  // ds_load_b128
    acc.x += v.x * sc;
    acc.y += v.y * sc;
    acc.z += v.z * sc;
    acc.w += v.w * sc;
  }
}

// ---------------------------------------------------------------------------
// Pull-mode aggregation: one wave per destination node, double-buffered
// async-to-LDS gather pipeline, zero feature atomics, one b128 store per lane.
// ---------------------------------------------------------------------------
__global__ __launch_bounds__(AGG_BLOCK) void aggregate_kernel(
    const float* __restrict__ src_emb, const float* __restrict__ dst_emb,
    const int* __restrict__ edge_src, const int* __restrict__ offsets,
    const float* __restrict__ scale_out, float* __restrict__ out,
    int n_nodes, int n_src) {
  __shared__ float stage[AGG_WAVES][2][CHUNK][GCN_D];  // 64 KB per block

  const int lane = threadIdx.x & 31;
  const int wslot = threadIdx.x >> 5;
  const int wave =
      __builtin_amdgcn_readfirstlane((int)((blockIdx.x * blockDim.x + threadIdx.x) >> 5));
  const int nwaves = (int)((gridDim.x * blockDim.x) >> 5);

  const float* sbuf0 = &stage[wslot][0][0][0];
  const float* sbuf1 = &stage[wslot][1][0][0];
  const unsigned lbase0 = (unsigned)(size_t)sbuf0 + (unsigned)(lane * 16);
  const unsigned lbase1 = (unsigned)(size_t)sbuf1 + (unsigned)(lane * 16);

  for (int d = wave; d < n_nodes; d += nwaves) {
    const int k0 = offsets[d];      // uniform -> s_load
    const int k1 = offsets[d + 1];  // uniform -> s_load
    const int K = k1 - k0;
    float4 acc = make_float4(0.f, 0.f, 0.f, 0.f);

    if (K > 0) {
      const int m0 = (K < CHUNK) ? K : CHUNK;
      issue_chunk(edge_src, k0, m0, src_emb, dst_emb, n_src, lbase0, lane);

      int buf = 0;
      for (int j = k0 + CHUNK; j < k1; j += CHUNK) {
        const int m = (k1 - j < CHUNK) ? (k1 - j) : CHUNK;
        issue_chunk(edge_src, j, m, src_emb, dst_emb, n_src,
                    (buf == 0) ? lbase1 : lbase0, lane);
        // in-order completion: <=m outstanding => previous chunk fully landed
        if (m == CHUNK) wait_async_le<CHUNK>(); else wait_async_le<0>();
        consume_chunk(edge_src, j - CHUNK, CHUNK, scale_out,
                      (buf == 0) ? sbuf0 : sbuf1, lane, acc);
        buf ^= 1;
      }

      wait_async_le<0>();
      const int last_j = k0 + ((K - 1) / CHUNK) * CHUNK;
      consume_chunk(edge_src, last_j, k1 - last_j, scale_out,
                    (buf == 0) ? sbuf0 : sbuf1, lane, acc);
    }

    const float sci = rsqrtf(fmaxf((float)K, 1.0f));  // indeg^-1/2
    float4 r = make_float4(acc.x * sci, acc.y * sci, acc.z * sci, acc.w * sci);
    *(float4*)(out + (size_t)d * GCN_D + lane * 4) = r;  // global_store_b128
  }
}

// ---------------------------------------------------------------------------
// zero int counters
// ---------------------------------------------------------------------------
__global__ __launch_bounds__(256) void zero_i32(int* __restrict__ p, int n) {
  int i = blockIdx.x * blockDim.x + threadIdx.x;
  int stride = gridDim.x * blockDim.x;
  for (; i < n; i += stride) p[i] = 0;
}

// ---------------------------------------------------------------------------
// integer degree histograms (no-return global_atomic_add_u32)
// ---------------------------------------------------------------------------
__global__ __launch_bounds__(256) void count_kernel(const int* __restrict__ src_idx,
                                                    const int* __restrict__ dst_idx,
                                                    int* __restrict__ cnt_out,
                                                    int* __restrict__ cnt_in, int E) {
  int i = blockIdx.x * blockDim.x + threadIdx.x;
  int stride = gridDim.x * blockDim.x;
  for (; i < E; i += stride) {
    atomicAdd(&cnt_out[src_idx[i]], 1);
    atomicAdd(&cnt_in[dst_idx[i]], 1);
  }
}

// ---------------------------------------------------------------------------
// single-block exclusive scan of in-degree counts -> offsets[n+1], cursor[n]
// ---------------------------------------------------------------------------
__global__ __launch_bounds__(1024) void scan_kernel(const int* __restrict__ counts,
                                                    int* __restrict__ offsets,
                                                    int* __restrict__ cursor, int n) {
  __shared__ int sh[1024];
  const int T = 1024;
  const int tid = threadIdx.x;
  const int chunk = (n + T - 1) / T;
  const int begin = tid * chunk;
  const int end = (begin + chunk < n) ? (begin + chunk) : n;

  int sum = 0;
  for (int i = begin; i < end; ++i) sum += counts[i];
  sh[tid] = sum;
  __syncthreads();

  for (int off = 1; off < T; off <<= 1) {
    int t = (tid >= off) ? sh[tid - off] : 0;
    __syncthreads();
    sh[tid] += t;
    __syncthreads();
  }
  int run = sh[tid] - sum;  // exclusive prefix of this thread's chunk

  for (int i = begin; i < end; ++i) {
    offsets[i] = run;
    cursor[i] = run;
    run += counts[i];
  }
  if (end >= n) offsets[n] = run;  // total == E (all writers agree)
}

// ---------------------------------------------------------------------------
// out-degree counts -> rsqrt(max(c,1)) as float
// ---------------------------------------------------------------------------
__global__ __launch_bounds__(256) void scale_out_kernel(const int* __restrict__ cnt,
                                                        float* __restrict__ scale, int n) {
  int i = blockIdx.x * blockDim.x + threadIdx.x;
  if (i < n) scale[i] = rsqrtf(fmaxf((float)cnt[i], 1.0f));
}

// ---------------------------------------------------------------------------
// bin edges by destination (CSR fill): returning atomics on 100K cursors
// ---------------------------------------------------------------------------
__global__ __launch_bounds__(256) void bin_kernel(const int* __restrict__ src_idx,
                                                  const int* __restrict__ dst_idx,
                                                  int* __restrict__ cursor,
                                                  int* __restrict__ edge_src, int E) {
  int i = blockIdx.x * blockDim.x + threadIdx.x;
  int stride = gridDim.x * blockDim.x;
  for (; i < E; i += stride) {
    int d = dst_idx[i];
    int pos = atomicAdd(&cursor[d], 1);
    edge_src[pos] = src_idx[i];
  }
}

// ---------------------------------------------------------------------------
// Launcher
// ---------------------------------------------------------------------------
extern "C" void kernel_launch(void* const* d_in, const int* in_sizes, int n_in,
                              void* d_out, int out_size, void* d_ws, size_t ws_size,
                              hipStream_t stream) {
  const float* src_emb = (const float*)d_in[0];
  const float* dst_emb = (const float*)d_in[1];
  const int* src_idx = (const int*)d_in[2];
  const int* dst_idx = (const int*)d_in[3];
  float* out = (float*)d_out;

  const int n_src = in_sizes[0] / GCN_D;  // 60000
  const int n_dst = in_sizes[1] / GCN_D;  // 40000
  const int n = n_src + n_dst;            // 100000 nodes
  const int E = in_sizes[2];              // 1600000 edges

  // workspace layout, ~ (5n+1+E)*4 bytes ~= 8.4 MB
  int* cnt_out = (int*)d_ws;                  // n
  int* cnt_in = cnt_out + n;                  // n
  int* offsets = cnt_in + n;                  // n+1
  int* cursor = offsets + (n + 1);            // n
  int* edge_src = cursor + n;                 // E
  float* scale_out = (float*)(edge_src + E);  // n

  zero_i32<<<512, 256, 0, stream>>>(cnt_out, 2 * n);  // cnt_out + cnt_in contiguous
  count_kernel<<<2048, 256, 0, stream>>>(src_idx, dst_idx, cnt_out, cnt_in, E);
  scan_kernel<<<1, 1024, 0, stream>>>(cnt_in, offsets, cursor, n);
  scale_out_kernel<<<(n + 255) / 256, 256, 0, stream>>>(cnt_out, scale_out, n);
  bin_kernel<<<2048, 256, 0, stream>>>(src_idx, dst_idx, cursor, edge_src, E);
  aggregate_kernel<<<4096, AGG_BLOCK, 0, stream>>>(src_emb, dst_emb, edge_src, offsets,
                                                   scale_out, out, n, n_src);
}